// Classifier_22058952032359
// MI455X (gfx1250) — compile-verified
//
#include <hip/hip_runtime.h>
#include <hip/hip_bf16.h>
#include <math.h>

// ---------- problem constants (match reference) ----------
#define NND   50000      // nodes
#define NED   500000     // edges
#define DIM   128        // feature dim
#define NG    100        // graphs
#define NC    5          // classes
#define NS    10         // n_support (= n_query)
#define HID   256        // relation hidden
#define NPG   (NND / NG) // nodes per graph = 500

typedef float v2f __attribute__((ext_vector_type(2)));
typedef float v8f __attribute__((ext_vector_type(8)));

// ---------- generic zero ----------
__global__ void zero_f4_kernel(float4* __restrict__ p, int n4) {
    int i = blockIdx.x * blockDim.x + threadIdx.x;
    if (i < n4) p[i] = make_float4(0.f, 0.f, 0.f, 0.f);
}

// ---------- degree / norm ----------
__global__ void deg_kernel(const int* __restrict__ dst, float* __restrict__ deg) {
    int e = blockIdx.x * blockDim.x + threadIdx.x;
    if (e < NED) atomicAdd(&deg[dst[e]], 1.0f);
}

__global__ void norm_kernel(float* __restrict__ norm) {
    int i = blockIdx.x * blockDim.x + threadIdx.x;
    if (i < NND) {
        float d = norm[i];
        norm[i] = rsqrtf(fmaxf(d, 1.0f));
    }
}

// ---------- h = feat * norm[row] ----------
__global__ void scale_kernel(const float4* __restrict__ in, const float* __restrict__ norm,
                             float4* __restrict__ out) {
    int i = blockIdx.x * blockDim.x + threadIdx.x;   // float4 index
    if (i < NND * DIM / 4) {
        int row = i >> 5;                            // i*4 / 128
        float s = norm[row];
        float4 v = in[i];
        v.x *= s; v.y *= s; v.z *= s; v.w *= s;
        out[i] = v;
    }
}

// ---------- edge aggregation: agg[dst] += hs[src] ----------
__global__ void aggregate_kernel(const float* __restrict__ hs,
                                 const int* __restrict__ src,
                                 const int* __restrict__ dst,
                                 float* __restrict__ agg) {
    int gid = blockIdx.x * blockDim.x + threadIdx.x; // E*32 threads
    int e = gid >> 5;
    if (e >= NED) return;
    int f = (gid & 31) << 2;                         // 4-float chunk
    int s = src[e], d = dst[e];
    float4 v = *(const float4*)(hs + s * DIM + f);
    float* p = agg + d * DIM + f;
    atomicAdd(p + 0, v.x);
    atomicAdd(p + 1, v.y);
    atomicAdd(p + 2, v.z);
    atomicAdd(p + 3, v.w);
}

// ---------- out = relu((agg @ W) * norm[row] + b[col])  via f32 WMMA ----------
// block = 256 threads = 8 waves, each wave owns one 16x16 column tile
// grid.x = NND/16 row tiles
__global__ void gcn_matmul_kernel(const float* __restrict__ agg,
                                  const float* __restrict__ W,
                                  const float* __restrict__ bias,
                                  const float* __restrict__ norm,
                                  float* __restrict__ out) {
    __shared__ float As[16 * DIM];                   // 8 KB A tile
    const int row0 = blockIdx.x * 16;
    const int tid  = threadIdx.x;

    // cooperative A-tile load: 2048 floats = 512 float4, 256 threads x 2
    const float4* asrc = (const float4*)(agg + row0 * DIM);
    float4* adst = (float4*)As;
    adst[tid]       = asrc[tid];
    adst[tid + 256] = asrc[tid + 256];
    __syncthreads();

    const int wave = tid >> 5;        // 0..7 -> column tile
    const int lane = tid & 31;
    const int half = lane >> 4;       // 0: lanes 0-15, 1: lanes 16-31
    const int lr   = lane & 15;
    const int col0 = wave * 16;
    const int colg = col0 + lr;

    v8f c = {};
    for (int kk = 0; kk < DIM / 4; ++kk) {
        const int k0 = kk * 4 + 2 * half;
        v2f a, b;
        a.x = As[lr * DIM + k0];
        a.y = As[lr * DIM + k0 + 1];
        b.x = W[k0 * DIM + colg];
        b.y = W[(k0 + 1) * DIM + colg];
        c = __builtin_amdgcn_wmma_f32_16x16x4_f32(
                /*neg_a=*/false, a, /*neg_b=*/false, b,
                /*c_mod=*/(short)0, c, /*reuse_a=*/false, /*reuse_b=*/false);
    }

    const float bc = bias[colg];
#pragma unroll
    for (int r = 0; r < 8; ++r) {
        int row = row0 + r + half * 8;               // C layout: M=r (+8 upper half)
        float v = c[r] * norm[row] + bc;
        out[row * DIM + colg] = fmaxf(v, 0.0f);
    }
}

// ---------- per-graph readout + linear: hsel[g] = h2[node] @ wl^T + bl ----------
__global__ void readout_kernel(const float* __restrict__ h2,
                               const int* __restrict__ to_fetch,
                               const float* __restrict__ wl,
                               const float* __restrict__ bl,
                               float* __restrict__ hsel) {
    const int g = blockIdx.x;        // 0..NG-1
    const int j = threadIdx.x;       // 0..127
    __shared__ float hs[DIM];
    const int node = to_fetch[g] + g * NPG;
    hs[j] = h2[node * DIM + j];
    __syncthreads();
    float acc = bl[j];
    for (int k = 0; k < DIM; ++k) acc += hs[k] * wl[j * DIM + k];
    hsel[g * DIM + j] = acc;
}

// ---------- support indices + prototypes ----------
__global__ void proto_kernel(const float* __restrict__ hsel,
                             const int* __restrict__ y_t,
                             int* __restrict__ sup,
                             float* __restrict__ proto,
                             float* __restrict__ proto_out) {
    __shared__ int s_sup[NC * NS];
    const int tid = threadIdx.x;     // 256 threads
    if (tid == 0) {
        int cnt[NC];
        for (int c = 0; c < NC; ++c) cnt[c] = 0;
        for (int i = 0; i < NG; ++i) {
            int c = y_t[i];
            if (c >= 0 && c < NC && cnt[c] < NS) {
                s_sup[c * NS + cnt[c]] = i;
                cnt[c]++;
            }
        }
    }
    __syncthreads();
    if (tid < NC * NS) sup[tid] = s_sup[tid];
    for (int idx = tid; idx < NC * DIM; idx += 256) {
        int c = idx / DIM, d = idx % DIM;
        float acc = 0.f;
        for (int k = 0; k < NS; ++k) acc += hsel[s_sup[c * NS + k] * DIM + d];
        float p = acc * (1.0f / NS);
        proto[idx] = p;
        proto_out[idx] = p;     // d_out[2..] : prototypes
    }
}

// ---------- relation head: one block per (query, class) pair ----------
__global__ void relation_kernel(const float* __restrict__ proto,
                                const float* __restrict__ hsel,
                                const int* __restrict__ sup,
                                const float* __restrict__ w1r,
                                const float* __restrict__ b1r,
                                const float* __restrict__ w2r,
                                const float* __restrict__ b2r,
                                float* __restrict__ scores) {
    const int p = blockIdx.x;        // pair = q*NC + c
    const int q = p / NC, c = p % NC;
    const int c0 = q / NS, k = q % NS;
    const int j = threadIdx.x;       // 0..255
    __shared__ float pair[2 * DIM];
    __shared__ float red[HID];
    const int qidx = sup[c0 * NS + k];
    if (j < DIM) pair[j] = proto[c * DIM + j];
    else         pair[j] = hsel[qidx * DIM + (j - DIM)];
    __syncthreads();
    float acc = b1r[j];
    for (int t = 0; t < 2 * DIM; ++t) acc += pair[t] * w1r[j * 2 * DIM + t];
    float hid = fmaxf(acc, 0.0f);
    red[j] = hid * w2r[j];
    __syncthreads();
    for (int s = HID / 2; s > 0; s >>= 1) {
        if (j < s) red[j] += red[j + s];
        __syncthreads();
    }
    if (j == 0) scores[p] = 1.0f / (1.0f + expf(-(red[0] + b2r[0])));
}

// ---------- loss + accuracy ----------
__global__ void loss_kernel(const float* __restrict__ scores, float* __restrict__ out) {
    const int tid = threadIdx.x;     // 256 threads
    __shared__ float red[256];
    // loss = sum over (5,10,5) of (score - eye(c0,c2))^2
    float v = 0.f;
    if (tid < NC * NS * NC) {
        int c2 = tid % NC;
        int c0 = tid / (NS * NC);
        float t = (c0 == c2) ? 1.0f : 0.0f;
        float d = scores[tid] - t;
        v = d * d;
    }
    red[tid] = v;
    __syncthreads();
    for (int s = 128; s > 0; s >>= 1) {
        if (tid < s) red[tid] += red[tid + s];
        __syncthreads();
    }
    if (tid == 0) out[0] = red[0];
    __syncthreads();
    // acc = mean over 50 queries of (argmax over 5 == c0)
    float a = 0.f;
    if (tid < NC * NS) {
        int c0 = tid / NS;
        const float* s5 = scores + tid * NC;
        int best = 0; float bv = s5[0];
        for (int c = 1; c < NC; ++c) { if (s5[c] > bv) { bv = s5[c]; best = c; } }
        a = (best == c0) ? 1.0f : 0.0f;
    }
    red[tid] = a;
    __syncthreads();
    for (int s = 128; s > 0; s >>= 1) {
        if (tid < s) red[tid] += red[tid + s];
        __syncthreads();
    }
    if (tid == 0) out[1] = red[0] * (1.0f / (NC * NS));
}

extern "C" void kernel_launch(void* const* d_in, const int* in_sizes, int n_in,
                              void* d_out, int out_size, void* d_ws, size_t ws_size,
                              hipStream_t stream) {
    const float* feat = (const float*)d_in[0];
    const float* W1   = (const float*)d_in[1];
    const float* b1   = (const float*)d_in[2];
    const float* W2   = (const float*)d_in[3];
    const float* b2   = (const float*)d_in[4];
    const float* wl   = (const float*)d_in[5];
    const float* bl   = (const float*)d_in[6];
    const float* w1r  = (const float*)d_in[7];
    const float* b1r  = (const float*)d_in[8];
    const float* w2r  = (const float*)d_in[9];
    const float* b2r  = (const float*)d_in[10];
    const int*   src  = (const int*)d_in[11];
    const int*   dst  = (const int*)d_in[12];
    const int*   tof  = (const int*)d_in[13];
    const int*   y_t  = (const int*)d_in[14];
    // d_in[15] = n_support (==NS, compile-time)

    float* out = (float*)d_out;

    // workspace layout (floats)
    float* ws     = (float*)d_ws;
    float* norm   = ws;                       // NND
    float* buf1   = norm + NND;               // NND*DIM
    float* buf2   = buf1 + (size_t)NND * DIM; // NND*DIM
    float* hsel   = buf2 + (size_t)NND * DIM; // NG*DIM
    float* proto  = hsel + NG * DIM;          // NC*DIM
    float* scores = proto + NC * DIM;         // 256 (pad)
    int*   sup    = (int*)(scores + 256);     // NC*NS

    const int ND4 = NND * DIM / 4;

    // 1) degree -> norm
    zero_f4_kernel<<<(NND / 4 + 255) / 256, 256, 0, stream>>>((float4*)norm, NND / 4);
    deg_kernel<<<(NED + 255) / 256, 256, 0, stream>>>(dst, norm);
    norm_kernel<<<(NND + 255) / 256, 256, 0, stream>>>(norm);

    // 2) layer 1
    scale_kernel<<<(ND4 + 255) / 256, 256, 0, stream>>>((const float4*)feat, norm, (float4*)buf1);
    zero_f4_kernel<<<(ND4 + 255) / 256, 256, 0, stream>>>((float4*)buf2, ND4);
    aggregate_kernel<<<NED * 32 / 256, 256, 0, stream>>>(buf1, src, dst, buf2);
    gcn_matmul_kernel<<<NND / 16, 256, 0, stream>>>(buf2, W1, b1, norm, buf1);

    // 3) layer 2
    scale_kernel<<<(ND4 + 255) / 256, 256, 0, stream>>>((const float4*)buf1, norm, (float4*)buf2);
    zero_f4_kernel<<<(ND4 + 255) / 256, 256, 0, stream>>>((float4*)buf1, ND4);
    aggregate_kernel<<<NED * 32 / 256, 256, 0, stream>>>(buf2, src, dst, buf1);
    gcn_matmul_kernel<<<NND / 16, 256, 0, stream>>>(buf1, W2, b2, norm, buf2);

    // 4) readout + linear
    readout_kernel<<<NG, DIM, 0, stream>>>(buf2, tof, wl, bl, hsel);

    // 5) prototypes (also writes d_out[2..641])
    proto_kernel<<<1, 256, 0, stream>>>(hsel, y_t, sup, proto, out + 2);

    // 6) relation head
    relation_kernel<<<NC * NS * NC, HID, 0, stream>>>(proto, hsel, sup,
                                                      w1r, b1r, w2r, b2r, scores);

    // 7) loss + acc -> d_out[0], d_out[1]
    loss_kernel<<<1, 256, 0, stream>>>(scores, out);
}